// Attention_24120536335036
// MI455X (gfx1250) — compile-verified
//
#include <hip/hip_runtime.h>
#include <math.h>

typedef __attribute__((ext_vector_type(2))) float v2f;
typedef __attribute__((ext_vector_type(8))) float v8f;

#define N_  4
#define L_  128
#define LM_ 256
#define D_  512
#define D2_ 1024
#define TL_ 4    // l-rows per score block

static __device__ __forceinline__ v8f wmma_f32_16x16x4(v2f a, v2f b, v8f c) {
  // 8 args: (neg_a, A, neg_b, B, c_mod, C, reuse_a, reuse_b)
  return __builtin_amdgcn_wmma_f32_16x16x4_f32(false, a, false, b, (short)0, c, false, false);
}

// Hardware tanh (TRANS32 op on gfx1250). Prefer the builtin (compiler models
// the TRANS hazard); else inline asm with an embedded v_nop per ISA 7.4:
// a TRANS result needs one independent op or v_nop before it is read.
static __device__ __forceinline__ float fast_tanh(float x) {
#if defined(__has_builtin) && __has_builtin(__builtin_amdgcn_tanhf)
  return __builtin_amdgcn_tanhf(x);
#else
  float y;
  asm("v_tanh_f32 %0, %1\n\tv_nop" : "=v"(y) : "v"(x));
  return y;
#endif
}

// ---------------------------------------------------------------------------
// Kernel 1a: q[r,e] = sum_d X[r,d] * W0[e,d] + b0[e]    (r = n*L+l, rows=512)
// One wave per 16x16 output tile; K-loop step 4 via V_WMMA_F32_16X16X4_F32.
// ---------------------------------------------------------------------------
__global__ __launch_bounds__(32) void k_qproj(const float* __restrict__ X,
                                              const float* __restrict__ W0,
                                              const float* __restrict__ b0,
                                              float* __restrict__ q) {
  const int lane = threadIdx.x;
  const int half = lane >> 4;
  const int lr   = lane & 15;
  const int nc = blockIdx.x * 16;   // e tile (cols)
  const int mr = blockIdx.y * 16;   // row tile

  const float* arow = X  + (size_t)(mr + lr) * D_;    // A[row][k]
  const float* bcol = W0 + (size_t)(nc + lr) * D2_;   // B[k][e] = W0[e*D2 + k]
  v8f acc = {};
  for (int k0 = 0; k0 < D_; k0 += 4) {
    const int kb = k0 + 2 * half;
    v2f a = *(const v2f*)(arow + kb);
    v2f b = *(const v2f*)(bcol + kb);
    acc = wmma_f32_16x16x4(a, b, acc);
  }
  const int col = nc + lr;
  const float bias = b0[col];
#pragma unroll
  for (int r = 0; r < 8; ++r) {
    const int row = mr + r + 8 * half;
    q[(size_t)row * D_ + col] = acc[r] + bias;
  }
}

// ---------------------------------------------------------------------------
// Kernel 1b: mpT[n][e][m] = sum_d W0[e, D+d] * memory[n][m][d]
// (memory-projection, stored TRANSPOSED so the score kernel is coalesced)
// ---------------------------------------------------------------------------
__global__ __launch_bounds__(32) void k_mproj(const float* __restrict__ mem,
                                              const float* __restrict__ W0,
                                              float* __restrict__ mpT) {
  const int lane = threadIdx.x;
  const int half = lane >> 4;
  const int lr   = lane & 15;
  const int mc = blockIdx.x * 16;   // m tile (cols)
  const int er = blockIdx.y * 16;   // e tile (rows)
  const int n  = blockIdx.z;

  const float* arow = W0 + (size_t)(er + lr) * D2_ + D_;                   // Wm[e][d]
  const float* bcol = mem + (size_t)n * LM_ * D_ + (size_t)(mc + lr) * D_; // B[k][m]=mem[n][m][k]
  v8f acc = {};
  for (int k0 = 0; k0 < D_; k0 += 4) {
    const int kb = k0 + 2 * half;
    v2f a = *(const v2f*)(arow + kb);
    v2f b = *(const v2f*)(bcol + kb);
    acc = wmma_f32_16x16x4(a, b, acc);
  }
  float* outp = mpT + (size_t)n * D_ * LM_;
  const int col = mc + lr;
#pragma unroll
  for (int r = 0; r < 8; ++r) {
    const int row = er + r + 8 * half;
    outp[(size_t)row * LM_ + col] = acc[r];
  }
}

// ---------------------------------------------------------------------------
// Kernel 2: scores + softmax (dominant: 67M hardware tanh). Block = 256
// threads (t=m), handles TL_=4 l-rows so each coalesced mpT load feeds
// 4 tanh-FMAs; all operand streams are L2-resident.
// ---------------------------------------------------------------------------
__global__ __launch_bounds__(256) void k_scores(const float* __restrict__ q,
                                                const float* __restrict__ mpT,
                                                const unsigned char* __restrict__ mask,
                                                const float* __restrict__ w1,
                                                const float* __restrict__ b1,
                                                float* __restrict__ attn) {
  __shared__ float qs[TL_][D_];
  __shared__ float w1s[D_];
  __shared__ float red[LM_];

  const int t = threadIdx.x;                       // m index
  const int n = blockIdx.x / (L_ / TL_);
  const int lbase = (blockIdx.x % (L_ / TL_)) * TL_;

  for (int i = t; i < D_; i += LM_) {
    w1s[i] = w1[i];
#pragma unroll
    for (int j = 0; j < TL_; ++j)
      qs[j][i] = q[(size_t)(n * L_ + lbase + j) * D_ + i];
  }
  __syncthreads();

  const float* mp = mpT + (size_t)n * D_ * LM_ + t;
  float s0 = 0.f, s1 = 0.f, s2 = 0.f, s3 = 0.f;
  for (int d = 0; d < D_; ++d) {
    const float mv = mp[(size_t)d * LM_];   // coalesced across the wave
    const float w  = w1s[d];
    s0 += w * fast_tanh(qs[0][d] + mv);
    s1 += w * fast_tanh(qs[1][d] + mv);
    s2 += w * fast_tanh(qs[2][d] + mv);
    s3 += w * fast_tanh(qs[3][d] + mv);
  }
  const float bb = b1[0];
  float s[TL_] = { s0 + bb, s1 + bb, s2 + bb, s3 + bb };

  const float NEG = -3.4028234663852886e38f;
#pragma unroll
  for (int j = 0; j < TL_; ++j)
    if (!mask[(size_t)(n * L_ + lbase + j) * LM_ + t]) s[j] = NEG;

  for (int j = 0; j < TL_; ++j) {
    red[t] = s[j];
    __syncthreads();
    for (int off = LM_ / 2; off > 0; off >>= 1) {
      if (t < off) red[t] = fmaxf(red[t], red[t + off]);
      __syncthreads();
    }
    const float mx = red[0];
    __syncthreads();
    const float e = __expf(s[j] - mx);
    red[t] = e;
    __syncthreads();
    for (int off = LM_ / 2; off > 0; off >>= 1) {
      if (t < off) red[t] += red[t + off];
      __syncthreads();
    }
    const float inv = 1.0f / red[0];
    __syncthreads();
    attn[(size_t)(n * L_ + lbase + j) * LM_ + t] = e * inv;
  }
}

// ---------------------------------------------------------------------------
// Kernel 3: out[n][l][d] = sum_m attn[n][l][m] * memory[n][m][d]  (WMMA)
// ---------------------------------------------------------------------------
__global__ __launch_bounds__(32) void k_out(const float* __restrict__ attn,
                                            const float* __restrict__ mem,
                                            float* __restrict__ out) {
  const int lane = threadIdx.x;
  const int half = lane >> 4;
  const int lr   = lane & 15;
  const int dc   = blockIdx.x * 16;  // d tile (cols)
  const int lt   = blockIdx.y * 16;  // l tile (rows)
  const int n    = blockIdx.z;

  const float* arow  = attn + (size_t)n * L_ * LM_ + (size_t)(lt + lr) * LM_;
  const float* bbase = mem + (size_t)n * LM_ * D_ + dc + lr;   // B[k][d]=mem[n][k][d]
  v8f acc = {};
  for (int k0 = 0; k0 < LM_; k0 += 4) {
    const int kb = k0 + 2 * half;
    v2f a = *(const v2f*)(arow + kb);
    v2f b;
    b.x = bbase[(size_t)kb * D_];
    b.y = bbase[(size_t)(kb + 1) * D_];
    acc = wmma_f32_16x16x4(a, b, acc);
  }
  const int col = dc + lr;
#pragma unroll
  for (int r = 0; r < 8; ++r) {
    const int row = lt + r + 8 * half;
    out[((size_t)n * L_ + row) * D_ + col] = acc[r];
  }
}

// ---------------------------------------------------------------------------
extern "C" void kernel_launch(void* const* d_in, const int* in_sizes, int n_in,
                              void* d_out, int out_size, void* d_ws, size_t ws_size,
                              hipStream_t stream) {
  const float* input_emb      = (const float*)d_in[0];
  const float* memory         = (const float*)d_in[1];
  const unsigned char* mask   = (const unsigned char*)d_in[2];
  const float* W0             = (const float*)d_in[3];
  const float* b0             = (const float*)d_in[4];
  const float* w1             = (const float*)d_in[5];
  const float* b1             = (const float*)d_in[6];
  float* out = (float*)d_out;

  // workspace: q (1 MB) | mpT (2 MB) | attn (0.5 MB)
  float* q    = (float*)d_ws;
  float* mpT  = q + (size_t)N_ * L_ * D_;
  float* attn = mpT + (size_t)N_ * D_ * LM_;

  dim3 g1(D_ / 16, (N_ * L_) / 16);           // 32 x 32 tiles
  k_qproj<<<g1, 32, 0, stream>>>(input_emb, W0, b0, q);

  dim3 g2(LM_ / 16, D_ / 16, N_);             // 16 x 32 x 4
  k_mproj<<<g2, 32, 0, stream>>>(memory, W0, mpT);

  k_scores<<<(N_ * L_) / TL_, LM_, 0, stream>>>(q, mpT, mask, w1, b1, attn);

  dim3 g3(D_ / 16, L_ / 16, N_);              // 32 x 8 x 4
  k_out<<<g3, 32, 0, stream>>>(attn, memory, out);
}